// SetCriterionOl_20873541058797
// MI455X (gfx1250) — compile-verified
//
#include <hip/hip_runtime.h>
#include <hip/hip_bf16.h>
#include <math.h>

#define EPSV   1e-7f
#define SDIM   1024
#define BATCH  32
#define GDIM   5
#define TOPK   5
#define NEGBIG (-1e30f)

typedef __attribute__((ext_vector_type(2))) float v2f;
typedef __attribute__((ext_vector_type(8))) float v8f;

__device__ __forceinline__ float sigm(float x) {
  return 1.0f / (1.0f + expf(-x));
}

// Insert (sc,id) into descending top-5 held in registers.
__device__ __forceinline__ void ins5(float sc, unsigned id,
    float& v0, float& v1, float& v2, float& v3, float& v4,
    unsigned& i0, unsigned& i1, unsigned& i2, unsigned& i3, unsigned& i4) {
  if (sc > v4) {
    v4 = sc; i4 = id;
    if (v4 > v3) { float tf=v3; v3=v4; v4=tf; unsigned tu=i3; i3=i4; i4=tu;
      if (v3 > v2) { tf=v2; v2=v3; v3=tf; tu=i2; i2=i3; i3=tu;
        if (v2 > v1) { tf=v1; v1=v2; v2=tf; tu=i1; i1=i2; i2=tu;
          if (v1 > v0) { tf=v0; v0=v1; v1=tf; tu=i0; i0=i1; i1=tu; } } } }
  }
}

// ---------------------------------------------------------------------------
// Kernel 1: per (batch, column-group) top-5 of the upper-triangular outer
// product sp[i]*ep[j]; 16x16 tiles via V_WMMA_F32_16X16X4_F32.
// Grid: (8 column groups, 32 batches), 256 threads (8 wave32).
// Block cg owns tile-columns tj = cg + 8k (k=0..7); wave w owns tile-rows
// ti ≡ w (mod 8) — no discarded loop iterations, scalar-only control flow
// around the WMMA (EXEC stays all-ones).
// ---------------------------------------------------------------------------
__global__ __launch_bounds__(256) void spans_topk_kernel(
    const float* __restrict__ fp,
    float* __restrict__ cand_v, unsigned* __restrict__ cand_i) {
  __shared__ float    s_sp[SDIM];
  __shared__ float    s_ep[128];
  __shared__ float    red_v[256 * TOPK];
  __shared__ unsigned red_i[256 * TOPK];

  const int cg   = blockIdx.x;
  const int b    = blockIdx.y;
  const int tid  = threadIdx.x;
  const int lane = tid & 31;
  const int wave = __builtin_amdgcn_readfirstlane(tid >> 5);  // uniform -> SGPR
  const int l16  = lane & 15;

  const float4* fp4 = (const float4*)(fp + (size_t)b * SDIM * 4);

  // start probs for all S
  for (int r = 0; r < 4; ++r) {
    int s = tid + r * 256;
    s_sp[s] = sigm(fp4[s].x);
  }
  // end probs for this group's 8 tile-columns, pre-scaled by 0.25 so the
  // 4-way K-replicated WMMA dot yields ~sp*ep directly.
  if (tid < 128) {
    int k = tid >> 4;
    int s = (cg + 8 * k) * 16 + (tid & 15);
    s_ep[tid] = 0.25f * sigm(fp4[s].z);
  }
  __syncthreads();

  // Preload this lane's 8 B operands (one per tile-column) into registers:
  // B[k][n] = 0.25*ep[n] replicated over all K rows (layout-robust).
  v2f bmat[8];
  #pragma unroll
  for (int k = 0; k < 8; ++k) {
    float bval = s_ep[k * 16 + l16];
    bmat[k] = (v2f){ bval, bval };
  }

  float    v0=NEGBIG, v1=NEGBIG, v2=NEGBIG, v3=NEGBIG, v4=NEGBIG;
  unsigned i0=0, i1=0, i2=0, i3=0, i4=0;

  for (int ti = wave; ti < 64; ti += 8) {
    const float aval = s_sp[ti * 16 + l16];       // A[m][k] = sp[m] for all k
    const v2f amat = (v2f){ aval, aval };
    const int mbase = ti * 16 + ((lane >> 4) << 3);  // C/D: M = v + 8*(lane>=16)
    #pragma unroll
    for (int k = 0; k < 8; ++k) {
      const int tj = cg + 8 * k;
      if (tj < ti) continue;                      // uniform: scalar branch
      v8f c = {0.f,0.f,0.f,0.f,0.f,0.f,0.f,0.f};
      v8f d = __builtin_amdgcn_wmma_f32_16x16x4_f32(
                  false, amat, false, bmat[k], (short)0, c, false, false);
      // tile-max prefilter: most tiles never touch the insert path
      float tmax = fmaxf(fmaxf(fmaxf(d[0],d[1]), fmaxf(d[2],d[3])),
                         fmaxf(fmaxf(d[4],d[5]), fmaxf(d[6],d[7])));
      if (tmax > v4) {
        const int n = tj * 16 + l16;
        #pragma unroll
        for (int e = 0; e < 8; ++e) {
          int m = mbase + e;
          float sc = d[e];
          if (m <= n && sc > v4) {
            unsigned id = ((unsigned)m << 10) | (unsigned)n;
            ins5(sc, id, v0,v1,v2,v3,v4, i0,i1,i2,i3,i4);
          }
        }
      }
    }
  }

  // block reduction of 256 top-5 lists -> 1 top-5 list
  red_v[tid*5+0]=v0; red_v[tid*5+1]=v1; red_v[tid*5+2]=v2; red_v[tid*5+3]=v3; red_v[tid*5+4]=v4;
  red_i[tid*5+0]=i0; red_i[tid*5+1]=i1; red_i[tid*5+2]=i2; red_i[tid*5+3]=i3; red_i[tid*5+4]=i4;
  __syncthreads();
  if (tid < 32) {
    float    w0=NEGBIG,w1=NEGBIG,w2=NEGBIG,w3=NEGBIG,w4=NEGBIG;
    unsigned j0=0,j1=0,j2=0,j3=0,j4=0;
    for (int src = tid; src < 256; src += 32)
      for (int q = 0; q < TOPK; ++q)
        ins5(red_v[src*5+q], red_i[src*5+q], w0,w1,w2,w3,w4, j0,j1,j2,j3,j4);
    red_v[tid*5+0]=w0; red_v[tid*5+1]=w1; red_v[tid*5+2]=w2; red_v[tid*5+3]=w3; red_v[tid*5+4]=w4;
    red_i[tid*5+0]=j0; red_i[tid*5+1]=j1; red_i[tid*5+2]=j2; red_i[tid*5+3]=j3; red_i[tid*5+4]=j4;
  }
  __syncthreads();
  if (tid == 0) {
    float    w0=NEGBIG,w1=NEGBIG,w2=NEGBIG,w3=NEGBIG,w4=NEGBIG;
    unsigned j0=0,j1=0,j2=0,j3=0,j4=0;
    for (int src = 0; src < 32; ++src)
      for (int q = 0; q < TOPK; ++q)
        ins5(red_v[src*5+q], red_i[src*5+q], w0,w1,w2,w3,w4, j0,j1,j2,j3,j4);
    const int base = (b * 8 + cg) * TOPK;
    cand_v[base+0]=w0; cand_v[base+1]=w1; cand_v[base+2]=w2; cand_v[base+3]=w3; cand_v[base+4]=w4;
    cand_i[base+0]=j0; cand_i[base+1]=j1; cand_i[base+2]=j2; cand_i[base+3]=j3; cand_i[base+4]=j4;
  }
}

// ---------------------------------------------------------------------------
// Kernel 2: per-batch merge of 8x5 candidates -> global top-5, exact conf
// recompute, GIoU vs gt windows, per-batch sum of squared errors.
// ---------------------------------------------------------------------------
__global__ void spans_final_kernel(
    const float* __restrict__ fp, const float* __restrict__ gtw,
    const int* __restrict__ sms,
    const float* __restrict__ cand_v, const unsigned* __restrict__ cand_i,
    float* __restrict__ ws_spans) {
  const int b = threadIdx.x;
  if (b >= BATCH) return;

  float    v0=NEGBIG,v1=NEGBIG,v2=NEGBIG,v3=NEGBIG,v4=NEGBIG;
  unsigned i0=0,i1=0,i2=0,i3=0,i4=0;
  for (int cg = 0; cg < 8; ++cg)
    for (int q = 0; q < TOPK; ++q) {
      const int base = (b * 8 + cg) * TOPK + q;
      ins5(cand_v[base], cand_i[base], v0,v1,v2,v3,v4, i0,i1,i2,i3,i4);
    }

  const float* fpb = fp + (size_t)b * SDIM * 4;
  const float smsf = (float)sms[b];
  unsigned ids[TOPK] = { i0, i1, i2, i3, i4 };

  float acc = 0.0f;
  for (int q = 0; q < TOPK; ++q) {
    const int st = (int)(ids[q] >> 10);
    const int ed = (int)(ids[q] & 1023u);
    // exact per-reference confidence: sigmoid(fp[st,0]) * sigmoid(fp[ed,2])
    const float conf = sigm(fpb[st*4+0]) * sigm(fpb[ed*4+2]);
    const float ps = (float)st + smsf;
    const float pe = (float)ed + smsf;
    float best = NEGBIG;
    for (int g = 0; g < GDIM; ++g) {
      const float gs = gtw[(b*GDIM+g)*2+0];
      const float ge = gtw[(b*GDIM+g)*2+1];
      float inter = fmaxf(fminf(pe, ge) - fmaxf(ps, gs), 0.0f);
      float uni   = (pe - ps) + (ge - gs) - inter;
      float iou   = inter / fmaxf(uni, EPSV);
      float enc   = fmaxf(fmaxf(pe, ge) - fminf(ps, gs), 0.0f);
      float giou  = iou - (enc - uni) / fmaxf(enc, EPSV);
      best = fmaxf(best, giou);
    }
    const float df = conf - best;
    acc += df * df;
  }
  ws_spans[b] = acc;
}

// ---------------------------------------------------------------------------
// Kernel 3: weighted BCE over 3 channels, block-reduced partial sums.
// ---------------------------------------------------------------------------
__device__ __forceinline__ float wbce(float x, float t) {
  float p = 1.0f / (1.0f + expf(-x));
  p = fminf(fmaxf(p, EPSV), 1.0f - EPSV);
  float bce = -(t * logf(p) + (1.0f - t) * logf(1.0f - p));
  float d = p - t;                    // GAMMA == 2 -> |p-t|^2 = (p-t)^2
  return bce * d * d;
}

__global__ __launch_bounds__(256) void labels_kernel(
    const float* __restrict__ fp,
    const float* __restrict__ t0, const float* __restrict__ t1,
    const float* __restrict__ t2, float* __restrict__ ws_lab) {
  __shared__ float sred[256];
  const int tid = threadIdx.x;
  const int blk = blockIdx.x;
  const float4* fp4 = (const float4*)fp;
  float acc = 0.0f;
  for (int r = 0; r < 4; ++r) {
    int e = blk * 1024 + r * 256 + tid;
    float4 q = fp4[e];
    acc += wbce(q.x, t0[e]) + wbce(q.y, t1[e]) + wbce(q.z, t2[e]);
  }
  sred[tid] = acc;
  __syncthreads();
  for (int s = 128; s > 0; s >>= 1) {
    if (tid < s) sred[tid] += sred[tid + s];
    __syncthreads();
  }
  if (tid == 0) ws_lab[blk] = sred[0];
}

// ---------------------------------------------------------------------------
// Kernel 4: deterministic fixed-order final sum.
// ---------------------------------------------------------------------------
__global__ void finalize_kernel(const float* __restrict__ ws_spans,
                                const float* __restrict__ ws_lab,
                                float* __restrict__ out) {
  float s1 = 0.0f, s2 = 0.0f;
  for (int i = 0; i < BATCH; ++i) s1 += ws_spans[i];
  for (int i = 0; i < BATCH; ++i) s2 += ws_lab[i];
  out[0] = s1 / (float)(BATCH * TOPK) + s2 / (float)(BATCH * SDIM);
}

extern "C" void kernel_launch(void* const* d_in, const int* in_sizes, int n_in,
                              void* d_out, int out_size, void* d_ws, size_t ws_size,
                              hipStream_t stream) {
  const float* fp  = (const float*)d_in[0];   // [32,1024,4]
  const float* t0  = (const float*)d_in[1];   // [32,1024]
  const float* t1  = (const float*)d_in[2];   // [32,1024]
  const float* t2  = (const float*)d_in[3];   // [32,1024]
  const float* gtw = (const float*)d_in[4];   // [32,5,2]
  const int*   sms = (const int*)d_in[5];     // [32]
  float* out = (float*)d_out;

  // workspace layout (floats): [0,1280) cand_v, [1280,2560) cand_i,
  // [2560,2592) spans partial, [2592,2624) labels partial
  float*    ws_f     = (float*)d_ws;
  float*    cand_v   = ws_f;
  unsigned* cand_i   = (unsigned*)(ws_f + 1280);
  float*    ws_spans = ws_f + 2560;
  float*    ws_lab   = ws_f + 2592;

  spans_topk_kernel<<<dim3(8, BATCH), 256, 0, stream>>>(fp, cand_v, cand_i);
  labels_kernel<<<BATCH, 256, 0, stream>>>(fp, t0, t1, t2, ws_lab);
  spans_final_kernel<<<1, 32, 0, stream>>>(fp, gtw, sms, cand_v, cand_i, ws_spans);
  finalize_kernel<<<1, 1, 0, stream>>>(ws_spans, ws_lab, out);
}